// SemiCRFModel_86294482912046
// MI455X (gfx1250) — compile-verified
//
#include <hip/hip_runtime.h>

#define LEN   4096
#define F     36
#define MSEG  100
#define ATTN  32
#define HID   64
#define TI    16                 // start positions per block
#define NP    (TI + MSEG)        // 116 prefix entries P[0..115]
#define NROWS (NP - 1)           // 115 x rows staged per block
#define PSTR  37                 // padded LDS row stride (coprime with 64 banks)
#define NEG_INFF (-1e9f)

typedef float v2f __attribute__((ext_vector_type(2)));
typedef float v8f __attribute__((ext_vector_type(8)));

__global__ __launch_bounds__(256)
void semicrf_score_kernel(const float* __restrict__ xp,
                          const float* __restrict__ wfp,
                          const float* __restrict__ qp,
                          const float* __restrict__ w1p,
                          const float* __restrict__ b1p,
                          const float* __restrict__ w2p,
                          const float* __restrict__ b2p,
                          float* __restrict__ outp) {
    __shared__ float XS[NROWS][F];     // staged x rows
    __shared__ float P [NP][PSTR];     // exclusive prefix of x
    __shared__ float P2[NP][PSTR];     // exclusive prefix of x*x
    __shared__ float W1[ATTN * HID];   // w1[a*64+h]
    __shared__ float B1W2[2 * HID];    // [0..63]=b1, [64..127]=w2
    __shared__ float WF[ATTN];
    __shared__ float sbuf[TI * MSEG];  // softmax context scalar per cell
    __shared__ float cscal;

    const int tid = threadIdx.x;
    const int i0  = blockIdx.x * TI;
    const float b2v = b2p[0];

    // CDNA5 prefetch path (lowers to global_prefetch_b8)
    __builtin_prefetch(xp + (size_t)i0 * F, 0, 1);
    __builtin_prefetch(w1p, 0, 1);

    // ---- stage x tile, weights into LDS ----
    for (int e = tid; e < NROWS * F; e += 256) {
        int row = e / F, f = e - row * F;
        int grow = i0 + row;
        XS[row][f] = (grow < LEN) ? xp[(size_t)grow * F + f] : 0.0f;
    }
    for (int e = tid; e < ATTN * HID; e += 256) W1[e] = w1p[e];
    if (tid < HID) { B1W2[tid] = b1p[tid]; B1W2[HID + tid] = w2p[tid]; }
    if (tid < ATTN) WF[tid] = wfp[tid];
    if (tid == 0) {
        float cc = 0.0f;
        #pragma unroll
        for (int a = 0; a < ATTN; a++) cc = fmaf(wfp[a], qp[a], cc);
        cscal = cc;
    }
    __syncthreads();

    // ---- block-local exclusive prefix sums (72 scanner threads) ----
    if (tid < 36) {
        int f = tid;
        float acc = 0.0f;
        for (int j = 0; j < NP; j++) {
            P[j][f] = acc;
            if (j < NROWS) acc += XS[j][f];
        }
    } else if (tid < 72) {
        int f = tid - 36;
        float acc = 0.0f;
        for (int j = 0; j < NP; j++) {
            P2[j][f] = acc;
            if (j < NROWS) { float v = XS[j][f]; acc = fmaf(v, v, acc); }
        }
    }
    __syncthreads();

    // ---- phase 1: per-cell softmax context scalar s ----
    const float c = cscal;
    for (int cid = tid; cid < TI * MSEG; cid += 256) {
        int di  = cid / MSEG;
        int lm1 = cid - di * MSEG;
        int l   = lm1 + 1;
        int i   = i0 + di;
        if (i + l > LEN) { sbuf[cid] = 0.0f; continue; }

        float lf = (float)l;
        float rl = 1.0f / lf;
        float aggm[F], aggv[F];
        #pragma unroll
        for (int f = 0; f < F; f++) {
            float s1 = P [di + l][f] - P [di][f];
            float s2 = P2[di + l][f] - P2[di][f];
            float mean = s1 * rl;
            aggm[f] = mean;
            aggv[f] = fmaf(-mean, mean, s2 * rl);
        }
        float a72 = lf * (1.0f / 500.0f);
        float a73 = __logf(lf + 1e-6f) * (1.0f / 6.0f);

        float mx = fmaxf(a72, a73), mn = fminf(a72, a73);
        #pragma unroll
        for (int f = 0; f < F; f++) {
            mx = fmaxf(mx, fmaxf(aggm[f], aggv[f]));
            mn = fminf(mn, fminf(aggm[f], aggv[f]));
        }
        float shift = (c >= 0.0f) ? c * mx : c * mn;  // = max(c*agg)

        float se = 0.0f, sae = 0.0f;
        #pragma unroll
        for (int f = 0; f < F; f++) {
            float e0 = __expf(fmaf(c, aggm[f], -shift));
            float e1 = __expf(fmaf(c, aggv[f], -shift));
            se += e0 + e1;
            sae = fmaf(aggm[f], e0, sae);
            sae = fmaf(aggv[f], e1, sae);
        }
        float e72 = __expf(fmaf(c, a72, -shift));
        float e73 = __expf(fmaf(c, a73, -shift));
        se += e72 + e73;
        sae = fmaf(a72, e72, sae);
        sae = fmaf(a73, e73, sae);

        sbuf[cid] = sae / se;
    }
    __syncthreads();

    // ---- phase 2: MLP via fp32 WMMA chains on the matrix pipe ----
    // H^T(64 x cells) = w1^T(64x32) @ ctx^T(32xcells), ctx^T[k][n] = s_n * wf[k]
    const int wave = tid >> 5;
    const int lane = tid & 31;
    const int half = lane >> 4;     // 0: lanes 0-15, 1: lanes 16-31
    const int lm   = lane & 15;

    const int ntile = (TI * MSEG) / 16;           // 100 tiles of 16 cells
    for (int tile = wave; tile < ntile; tile += 8) {
        const int cellbase = tile * 16;
        const float s = sbuf[cellbase + lm];      // cell n = lm (both halves)

        v8f acc[4] = {};                          // 4 hidden tiles of 16
        #pragma unroll
        for (int k0 = 0; k0 < ATTN; k0 += 4) {
            const int ka = k0 + (half ? 2 : 0);   // K slots: lanes0-15->K0,K1; 16-31->K2,K3
            v2f B;                                // B(4x16): B[k][n] = s_n * wf[k]
            B.x = s * WF[ka];
            B.y = s * WF[ka + 1];
            #pragma unroll
            for (int ht = 0; ht < 4; ht++) {
                v2f A;                            // A(16x4): A[m][k] = w1[k][ht*16+m]
                A.x = W1[ka * HID + ht * 16 + lm];
                A.y = W1[(ka + 1) * HID + ht * 16 + lm];
                acc[ht] = __builtin_amdgcn_wmma_f32_16x16x4_f32(
                    false, A, false, B, (short)0, acc[ht], false, false);
            }
        }

        // epilogue: D layout -> lane holds cell n=lm; VGPR r holds h = ht*16 + r + half*8
        float p = 0.0f;
        #pragma unroll
        for (int ht = 0; ht < 4; ht++) {
            #pragma unroll
            for (int r = 0; r < 8; r++) {
                int h = ht * 16 + r + half * 8;
                float hv = fmaxf(acc[ht][r] + B1W2[h], 0.0f);
                p = fmaf(hv, B1W2[HID + h], p);
            }
        }
        p += __shfl_xor(p, 16, 32);               // combine the two half-lane row groups

        if (half == 0) {
            int cell = cellbase + lm;
            int di   = cell / MSEG;
            int lm1  = cell - di * MSEG;
            int i    = i0 + di;
            int l    = lm1 + 1;
            float res = (i + l <= LEN) ? (p + b2v) : NEG_INFF;
            outp[(size_t)i * MSEG + lm1] = res;
        }
    }
}

extern "C" void kernel_launch(void* const* d_in, const int* in_sizes, int n_in,
                              void* d_out, int out_size, void* d_ws, size_t ws_size,
                              hipStream_t stream) {
    (void)in_sizes; (void)n_in; (void)d_ws; (void)ws_size; (void)out_size;
    const float* x      = (const float*)d_in[0];
    const float* w_feat = (const float*)d_in[1];
    const float* query  = (const float*)d_in[2];
    const float* w1     = (const float*)d_in[3];
    const float* b1     = (const float*)d_in[4];
    const float* w2     = (const float*)d_in[5];
    const float* b2     = (const float*)d_in[6];
    float* out          = (float*)d_out;

    dim3 grid(LEN / TI);   // 256 blocks
    dim3 block(256);       // 8 wave32s
    semicrf_score_kernel<<<grid, block, 0, stream>>>(x, w_feat, query, w1, b1, w2, b2, out);
}